// BasicVoxelization_61040075210987
// MI455X (gfx1250) — compile-verified
//
#include <hip/hip_runtime.h>
#include <stdint.h>

// ---- Problem constants (match reference) ----
#define GRID_X 704
#define GRID_Y 800
#define GRID_Z 20
#define BIG    (GRID_X * GRID_Y * GRID_Z)   // 11,264,000
#define MAX_PTS 10
#define MAX_VOX 40000
#define NBATCH  8
#define SLOTS   (MAX_VOX * MAX_PTS)         // 400,000

#define PC_MIN_X 0.0f
#define PC_MIN_Y (-40.0f)
#define PC_MIN_Z (-3.0f)
#define VS_X 0.1f
#define VS_Y 0.1f
#define VS_Z 0.2f

// Output region sizes (float elements)
#define OUT_VOX  ((size_t)NBATCH * MAX_VOX * MAX_PTS * 4)  // 12,800,000
#define OUT_NUM  ((size_t)NBATCH * MAX_VOX)                // 320,000
#define OUT_COOR ((size_t)NBATCH * MAX_VOX * 4)            // 1,280,000
#define OUT_TOT  (OUT_VOX + OUT_NUM + OUT_COOR + 3)

#define OUT_ZERO4 ((OUT_VOX + OUT_NUM) / 4)  // 3,280,000 float4 of zeros
#define OUT_COOR4 (OUT_COOR / 4)             // 320,000 coor rows

#define INTMAX32 0x7fffffff
#define IDXMASK  0x00ffffff

// ---------- init: outputs (vectorized b128 stores) ----------
__global__ void k_init_out(float4* __restrict__ out4) {
    size_t idx = (size_t)blockIdx.x * blockDim.x + threadIdx.x;
    size_t stride = (size_t)gridDim.x * blockDim.x;
    size_t total4 = OUT_ZERO4 + OUT_COOR4;
    for (size_t i = idx; i < total4; i += stride) {
        float4 v;
        if (i < OUT_ZERO4) {
            v = make_float4(0.0f, 0.0f, 0.0f, 0.0f);       // voxels + num_points
        } else {
            int row = (int)(i - OUT_ZERO4);                 // coor row
            v = make_float4((float)(row / MAX_VOX), -1.0f, -1.0f, -1.0f);
        }
        out4[i] = v;
    }
    if (idx == 0) {                                         // grid_size tail
        float* out = (float*)out4;
        out[OUT_TOT - 3] = (float)GRID_X;
        out[OUT_TOT - 2] = (float)GRID_Y;
        out[OUT_TOT - 1] = (float)GRID_Z;
    }
}

// ---------- init: map + slot buffer (contiguous, vectorized) ----------
__global__ void k_init_ws(int4* __restrict__ p) {
    size_t idx = (size_t)blockIdx.x * blockDim.x + threadIdx.x;
    size_t stride = (size_t)gridDim.x * blockDim.x;
    size_t total4 = ((size_t)BIG + SLOTS) / 4;              // 2,916,000
    int4 v = make_int4(INTMAX32, INTMAX32, INTMAX32, INTMAX32);
    for (size_t i = idx; i < total4; i += stride) p[i] = v;
}

// ---------- pass 0: lid per point + tagged first-point key per voxel ----------
// key = tag | i, with tag = (7-b)<<24: later batches have smaller tags, so
// atomicMin always beats stale entries -> no map reset needed between batches.
__global__ void k_first_min(const float* __restrict__ pb, int N,
                            int* __restrict__ map0, int* __restrict__ lidA, int tag) {
    int i = blockIdx.x * blockDim.x + threadIdx.x;
    if (i >= N) return;
    const float* p5 = pb + (size_t)i * 5;
    float x = p5[1], y = p5[2], z = p5[3];
    int cx = (int)floorf((x - PC_MIN_X) / VS_X);
    int cy = (int)floorf((y - PC_MIN_Y) / VS_Y);
    int cz = (int)floorf((z - PC_MIN_Z) / VS_Z);
    bool valid = (cx >= 0) && (cx < GRID_X) && (cy >= 0) && (cy < GRID_Y) &&
                 (cz >= 0) && (cz < GRID_Z);
    int lid = (cz * GRID_Y + cy) * GRID_X + cx;
    lidA[i] = valid ? lid : -1;
    if (valid) atomicMin(&map0[lid], tag | i);
}

// ---------- scan stage 1: creator flags + per-block sums ----------
__global__ void k_scan1(const int* __restrict__ lidA, int N,
                        const int* __restrict__ map0,
                        int* __restrict__ flags, int* __restrict__ bsum, int tag) {
    __shared__ int red[256];
    int tid = threadIdx.x;
    int i = blockIdx.x * 256 + tid;
    int f = 0;
    if (i < N) {
        int L = lidA[i];
        f = (L >= 0 && map0[L] == (tag | i)) ? 1 : 0;
        flags[i] = f;
    }
    red[tid] = f;
    __syncthreads();
    for (int s = 128; s > 0; s >>= 1) {
        if (tid < s) red[tid] += red[tid + s];
        __syncthreads();
    }
    if (tid == 0) bsum[blockIdx.x] = red[0];
}

// ---------- scan stage 2: scan of block sums (single block, nblk <= 1024) ----------
__global__ void k_scan2(const int* __restrict__ bsum, int nblk, int* __restrict__ boff) {
    __shared__ int sc[1024];
    int t = threadIdx.x;
    int v = (t < nblk) ? bsum[t] : 0;
    sc[t] = v;
    __syncthreads();
    for (int off = 1; off < 1024; off <<= 1) {
        int a = (t >= off) ? sc[t - off] : 0;
        __syncthreads();
        sc[t] += a;
        __syncthreads();
    }
    if (t < nblk) boff[t] = sc[t] - v;  // exclusive
}

// ---------- scan stage 3: intra-block exclusive scan -> rankExcl ----------
// Stages flags via the CDNA5 async tensor-memory path (ASYNCcnt).
__global__ void k_scan3(const int* __restrict__ flags, const int* __restrict__ boff,
                        int* __restrict__ rank, int N) {
    __shared__ int sflag[256];
    __shared__ int tmp[256];
    int tid = threadIdx.x;
    int i = blockIdx.x * 256 + tid;
    if (i < N) {
        unsigned loff = (unsigned)(uintptr_t)&sflag[tid];  // LDS byte offset
        unsigned goff = (unsigned)i * 4u;                  // byte offset from base
        // GVS mode: global base in SGPR plus VGPR byte offset, LDS address in VGPR; tracked by ASYNCcnt
        asm volatile("global_load_async_to_lds_b32 %0, %1, %2"
                     :: "v"(loff), "v"(goff), "s"(flags) : "memory");
    } else {
        sflag[tid] = 0;
    }
    asm volatile("s_wait_asynccnt 0" ::: "memory");
    __syncthreads();
    int v = sflag[tid];
    tmp[tid] = v;
    __syncthreads();
    for (int off = 1; off < 256; off <<= 1) {
        int a = (tid >= off) ? tmp[tid - off] : 0;
        __syncthreads();
        tmp[tid] += a;
        __syncthreads();
    }
    if (i < N) rank[i] = boff[blockIdx.x] + (tmp[tid] - v);  // exclusive rank
}

// ---------- resolve slot, write coors, cascade-insert point index ----------
__global__ void k_resolve_insert(int N, const int* __restrict__ map0,
                                 const int* __restrict__ rank,
                                 const int* __restrict__ lidA,
                                 int* __restrict__ slotbuf,
                                 float* __restrict__ coorb, float bval) {
    int i = blockIdx.x * blockDim.x + threadIdx.x;
    if (i >= N) return;
    int L = lidA[i];
    if (L < 0) return;                       // invalid point
    int i0 = map0[L] & IDXMASK;              // creator point index (tag stripped;
                                             // prefix == current tag by construction)
    int s = rank[i0];                        // slot = rank of creator among creators
    if (s >= MAX_VOX) return;                // dropped voxel
    if (i0 == i) {
        int cz = L / (GRID_X * GRID_Y);
        int rem = L - cz * (GRID_X * GRID_Y);
        int cy = rem / GRID_X;
        int cx = rem - cy * GRID_X;
        float* cr = coorb + (size_t)s * 4;
        cr[0] = bval;
        cr[1] = (float)cz;
        cr[2] = (float)cy;
        cr[3] = (float)cx;
    }
    // Concurrent insertion: slots end up holding the MAX_PTS smallest point
    // indices of this voxel in ascending order (== reference's stable order).
    int* slots = slotbuf + (size_t)s * MAX_PTS;
    int carry = i;
    #pragma unroll
    for (int t = 0; t < MAX_PTS; ++t) {
        if (carry == INTMAX32) break;
        int old = atomicMin(&slots[t], carry);
        carry = (old > carry) ? old : carry;  // displaced value moves on
    }
}

// ---------- finalize: gather point features per voxel, write num_points ----------
__global__ void k_finalize(const float* __restrict__ pb, int* __restrict__ slotbuf,
                           float* __restrict__ voxb, float* __restrict__ numb) {
    int v = blockIdx.x * blockDim.x + threadIdx.x;
    if (v >= MAX_VOX) return;
    int* slots = slotbuf + (size_t)v * MAX_PTS;
    float* vbase = voxb + (size_t)v * MAX_PTS * 4;
    int cnt = 0;
    #pragma unroll
    for (int s = 0; s < MAX_PTS; ++s) {
        int pi = slots[s];
        if (pi == INTMAX32) break;           // slots fill contiguously from 0
        const float* p5 = pb + (size_t)pi * 5;
        float* vrow = vbase + (size_t)s * 4;
        vrow[0] = p5[1];
        vrow[1] = p5[2];
        vrow[2] = p5[3];
        vrow[3] = p5[4];
        slots[s] = INTMAX32;                 // self-clean for next batch
        ++cnt;
    }
    if (cnt > 0) numb[v] = (float)cnt;       // empty voxels stay at init value 0
}

extern "C" void kernel_launch(void* const* d_in, const int* in_sizes, int n_in,
                              void* d_out, int out_size, void* d_ws, size_t ws_size,
                              hipStream_t stream) {
    const float* pts = (const float*)d_in[0];
    int n_total = in_sizes[0] / 5;
    int N = n_total / NBATCH;            // 120000
    int nblk = (N + 255) / 256;          // 469

    // workspace layout (ints): [map0 | slotbuf | flags | rank | lidA | bsum | boff]
    int* map0    = (int*)d_ws;
    int* slotbuf = map0 + BIG;
    int* flags   = slotbuf + SLOTS;
    int* rank    = flags + N;
    int* lidA    = rank + N;
    int* bsum    = lidA + N;
    int* boff    = bsum + 1024;

    float* out = (float*)d_out;
    float* outV = out;
    float* outN = out + OUT_VOX;
    float* outC = out + OUT_VOX + OUT_NUM;

    k_init_out<<<8192, 256, 0, stream>>>((float4*)out);
    k_init_ws<<<8192, 256, 0, stream>>>((int4*)map0);   // covers map0 + slotbuf

    int fblk = (MAX_VOX + 255) / 256;

    for (int b = 0; b < NBATCH; ++b) {
        const float* pb = pts + (size_t)b * N * 5;
        float* voxb  = outV + (size_t)b * MAX_VOX * MAX_PTS * 4;
        float* numb  = outN + (size_t)b * MAX_VOX;
        float* coorb = outC + (size_t)b * MAX_VOX * 4;
        int tag = (NBATCH - 1 - b) << 24;

        k_first_min<<<nblk, 256, 0, stream>>>(pb, N, map0, lidA, tag);
        k_scan1<<<nblk, 256, 0, stream>>>(lidA, N, map0, flags, bsum, tag);
        k_scan2<<<1, 1024, 0, stream>>>(bsum, nblk, boff);
        k_scan3<<<nblk, 256, 0, stream>>>(flags, boff, rank, N);
        k_resolve_insert<<<nblk, 256, 0, stream>>>(N, map0, rank, lidA, slotbuf,
                                                   coorb, (float)b);
        k_finalize<<<fblk, 256, 0, stream>>>(pb, slotbuf, voxb, numb);
    }
}